// LinearLossNaive_83365315215380
// MI455X (gfx1250) — compile-verified
//
#include <hip/hip_runtime.h>

// Problem constants from the reference
#define C_CH   128
#define HW     (512 * 512)          // spatial elements per channel
#define BPC    64                   // blocks per channel
#define EPB    (HW / BPC)           // 4096 elements per block per channel
#define THREADS 256                 // 8 wave32 per block
#define F4_PER_THREAD (EPB / (THREADS * 4))  // 4 float4 loads per tensor per thread

typedef __attribute__((ext_vector_type(2))) float v2f;
typedef __attribute__((ext_vector_type(4))) float v4f;
typedef __attribute__((ext_vector_type(8))) float v8f;

// Kernel 1: streaming per-channel partial sums of (input - target).
// Grid: (BPC, C_CH). Each block reduces 4096 contiguous floats of one channel
// (from both tensors) and writes one partial to ws[c*BPC + b].
// Loads are non-temporal: 268 MB single-pass stream > 192 MB L2, zero reuse.
__global__ __launch_bounds__(THREADS)
void channel_diff_partials(const float* __restrict__ in,
                           const float* __restrict__ tg,
                           float* __restrict__ partials) {
    const int b = blockIdx.x;
    const int c = blockIdx.y;
    const long base = (long)c * HW + (long)b * EPB;   // multiple of 4 -> float4 aligned
    const v4f* __restrict__ in4 = (const v4f*)(in + base);
    const v4f* __restrict__ tg4 = (const v4f*)(tg + base);
    const int t = threadIdx.x;

    float a0 = 0.f, a1 = 0.f, a2 = 0.f, a3 = 0.f;
#pragma unroll
    for (int i = 0; i < F4_PER_THREAD; ++i) {
        v4f x = __builtin_nontemporal_load(&in4[t + i * THREADS]);  // global_load_b128 th:NT
        v4f y = __builtin_nontemporal_load(&tg4[t + i * THREADS]);  // global_load_b128 th:NT
        a0 += x[0] - y[0];
        a1 += x[1] - y[1];
        a2 += x[2] - y[2];
        a3 += x[3] - y[3];
    }

    // Cross-lane wave reduction via the matrix pipe:
    // A = per-lane accumulator pair (16x4 f32 across the wave), B = ones(4x16),
    // D[m][n] = rowsum(m). Full wave sum = sum over all 16 rows.
    v2f a;   a.x = a0 + a2;  a.y = a1 + a3;
    v2f one; one.x = 1.0f;   one.y = 1.0f;
    v8f acc = {};
    acc = __builtin_amdgcn_wmma_f32_16x16x4_f32(
        /*neg_a=*/false, a, /*neg_b=*/false, one,
        /*c_mod=*/(short)0, acc, /*reuse_a=*/false, /*reuse_b=*/false);

    // Lanes 0-15 hold rows 0-7 in acc[0..7]; lanes 16-31 hold rows 8-15.
    float s = acc[0] + acc[1] + acc[2] + acc[3] + acc[4] + acc[5] + acc[6] + acc[7];
    float wave_total = s + __shfl_xor(s, 16, 32);   // fold the two half-waves

    __shared__ float wsum[THREADS / 32];
    const int lane = t & 31, wid = t >> 5;
    if (lane == 0) wsum[wid] = wave_total;
    __syncthreads();
    if (t == 0) {
        float tot = 0.f;
#pragma unroll
        for (int w = 0; w < THREADS / 32; ++w) tot += wsum[w];   // fixed order
        partials[c * BPC + b] = tot;
    }
}

// Kernel 2: deterministic finish. d[c] = sum of that channel's 64 partials in
// fixed order; loss = sum_c d[c]^2 / 16384.
__global__ __launch_bounds__(C_CH)
void finish_loss(const float* __restrict__ partials, float* __restrict__ out) {
    __shared__ float sm[C_CH];
    const int c = threadIdx.x;   // 128 threads, one per channel
    float d = 0.f;
    for (int b = 0; b < BPC; ++b) d += partials[c * BPC + b];
    sm[c] = d * d;
    __syncthreads();
    for (int s = C_CH / 2; s > 0; s >>= 1) {
        if (c < s) sm[c] += sm[c + s];
        __syncthreads();
    }
    if (c == 0) out[0] = sm[0] * (1.0f / (128.0f * 128.0f));
}

extern "C" void kernel_launch(void* const* d_in, const int* in_sizes, int n_in,
                              void* d_out, int out_size, void* d_ws, size_t ws_size,
                              hipStream_t stream) {
    const float* in = (const float*)d_in[0];
    const float* tg = (const float*)d_in[1];
    float* ws  = (float*)d_ws;      // needs BPC*C_CH*4 = 32 KB
    float* out = (float*)d_out;

    dim3 grid(BPC, C_CH);           // 8192 blocks
    channel_diff_partials<<<grid, THREADS, 0, stream>>>(in, tg, ws);
    finish_loss<<<1, C_CH, 0, stream>>>(ws, out);
}